// SpatialPyramidPool1d_2869038153932
// MI455X (gfx1250) — compile-verified
//
#include <hip/hip_runtime.h>
#include <hip/hip_bf16.h>

#define BB 32
#define CC 256
#define LL 8192
#define NUM_LEVELS 6
#define ROWS_PER_BLOCK 4            // one wave32 per row
#define THREADS (ROWS_PER_BLOCK * 32)
#define OUT_PER_BATCH (CC * 63)     // sum_{i<6} C * 2^i = 256*63 = 16128

#if __has_builtin(__builtin_amdgcn_global_load_async_to_lds_b128) && \
    __has_builtin(__builtin_amdgcn_s_wait_asynccnt)
#define USE_ASYNC_LDS 1
#else
#define USE_ASYNC_LDS 0
#endif

typedef int v4i __attribute__((ext_vector_type(4)));
typedef __attribute__((address_space(1))) v4i gv4i_t;   // global 128-bit vector
typedef __attribute__((address_space(3))) v4i lv4i_t;   // LDS 128-bit vector

__global__ __launch_bounds__(THREADS, 1)
void spp1d_kernel(const float* __restrict__ x,
                  const int* __restrict__ orig_len,
                  float* __restrict__ out) {
    extern __shared__ float smem[];   // ROWS_PER_BLOCK * LL floats = 128 KB

    const int w    = threadIdx.x >> 5;   // wave id within block
    const int lane = threadIdx.x & 31;

    // block -> (batch, 4 consecutive channels)
    const int blocksPerBatch = CC / ROWS_PER_BLOCK;      // 64
    const int b = blockIdx.x / blocksPerBatch;
    const int c = (blockIdx.x % blocksPerBatch) * ROWS_PER_BLOCK + w;

    const float* grow = x + ((size_t)b * CC + c) * (size_t)LL;
    float*       lrow = smem + (size_t)w * LL;

    // ---- Stage this wave's row (32 KB) into LDS --------------------------
    const float4* g4 = reinterpret_cast<const float4*>(grow);
    float4*       l4 = reinterpret_cast<float4*>(lrow);
#if USE_ASYNC_LDS
    #pragma unroll 4
    for (int q = 0; q < (LL / 4) / 32; ++q) {            // 64 async B128 ops
        const int idx = lane + 32 * q;
        __builtin_amdgcn_global_load_async_to_lds_b128(
            (gv4i_t*)(g4 + idx), (lv4i_t*)(l4 + idx), 0, 0);
    }
    __builtin_amdgcn_s_wait_asynccnt(0);
#else
    #pragma unroll 4
    for (int q = 0; q < (LL / 4) / 32; ++q) {
        const int idx = lane + 32 * q;
        l4[idx] = g4[idx];
    }
#endif
    __syncthreads();

    // ---- Compute 63 pyramid bins for this row from LDS -------------------
    const int n = orig_len[b];                 // SHIFT == 0
    const size_t outb = (size_t)b * OUT_PER_BATCH;

    for (int i = 0; i < NUM_LEVELS; ++i) {
        const int nb = 1 << i;
        const int fl = n >> i;                       // n / nb
        const int ce = (n + nb - 1) >> i;            // ceil(n / nb)
        const size_t obase = outb + (size_t)CC * (nb - 1) + (size_t)c * nb;

        for (int j = 0; j < nb; ++j) {
            const int s = j * fl;
            const int e = s + ce;                    // always <= n <= LL
            float m = -__builtin_inff();
            for (int p = s + lane; p < e; p += 32)
                m = fmaxf(m, lrow[p]);
            // wave32 max-reduction
            m = fmaxf(m, __shfl_xor(m, 16, 32));
            m = fmaxf(m, __shfl_xor(m,  8, 32));
            m = fmaxf(m, __shfl_xor(m,  4, 32));
            m = fmaxf(m, __shfl_xor(m,  2, 32));
            m = fmaxf(m, __shfl_xor(m,  1, 32));
            if (lane == 0) out[obase + j] = m;
        }
    }
}

extern "C" void kernel_launch(void* const* d_in, const int* in_sizes, int n_in,
                              void* d_out, int out_size, void* d_ws, size_t ws_size,
                              hipStream_t stream) {
    (void)in_sizes; (void)n_in; (void)out_size; (void)d_ws; (void)ws_size;
    const float* x        = (const float*)d_in[0];
    const int*   orig_len = (const int*)d_in[1];
    float*       out      = (float*)d_out;

    const size_t shmem = (size_t)ROWS_PER_BLOCK * LL * sizeof(float);  // 128 KB
    static_assert(ROWS_PER_BLOCK * LL * sizeof(float) == 131072, "lds size");

    // Opt in to >64KB dynamic LDS (no-op if already permitted).
    (void)hipFuncSetAttribute((const void*)spp1d_kernel,
                              hipFuncAttributeMaxDynamicSharedMemorySize,
                              (int)shmem);

    dim3 grid(BB * CC / ROWS_PER_BLOCK);   // 2048 blocks
    dim3 block(THREADS);                   // 128 threads = 4 waves
    spp1d_kernel<<<grid, block, shmem, stream>>>(x, orig_len, out);
}